// LightSpeedMemory_21638045237632
// MI455X (gfx1250) — compile-verified
//
#include <hip/hip_runtime.h>
#include <math.h>

#define LN 4096
#define DN 256
#define NEG_BIG (-1000000000.0f)
#define NT 4   // N-tiles (16 cols each) per wave in attend: A-fragment reuse x4

typedef __attribute__((ext_vector_type(2))) float v2f;
typedef __attribute__((ext_vector_type(8))) float v8f;

// ---------------------------------------------------------------------------
// Kernel 1: per-token scalars: photon flag, v = (1-mass)*c, photon_mask output
// ---------------------------------------------------------------------------
__global__ void prep_kernel(const int* __restrict__ pos_indices,
                            const float* __restrict__ particle_mass,
                            const float* __restrict__ log_c,
                            int* __restrict__ photon,
                            float* __restrict__ v_arr,
                            float* __restrict__ photon_mask) {
    int j = blockIdx.x * blockDim.x + threadIdx.x;
    if (j >= LN) return;
    float c = expf(log_c[0]);
    int idx = pos_indices[j] % 10;
    float pm = particle_mass[idx];
    float mass = 1.0f / (1.0f + expf(-pm));
    int ph = (mass < 0.01f) ? 1 : 0;
    photon[j] = ph;
    v_arr[j] = (1.0f - mass) * c;
    photon_mask[j] = ph ? 1.0f : 0.0f;
}

// ---------------------------------------------------------------------------
// Kernel 2: scores matrix + per-row softmax stats (rowmax / rowsum).
// One 256-thread block per row i; thread t handles j = chunk*256 + t.
// Output-critical exp (redshift) uses libm expf; softmax-sum uses __expf
// (consistent with the weights recomputed in attend).
// ---------------------------------------------------------------------------
__global__ void scores_kernel(const float* __restrict__ base,
                              const int* __restrict__ token_positions,
                              const int* __restrict__ photon,
                              const float* __restrict__ v_arr,
                              const float* __restrict__ log_c,
                              float* __restrict__ scores,
                              float* __restrict__ rowmax,
                              float* __restrict__ rowsum) {
    __shared__ float red[8];
    const int i = blockIdx.x;
    const int t = threadIdx.x;
    const float c = expf(log_c[0]);
    const float pos_i = (float)token_positions[i];
    const int ph_i = photon[i];

    float vals[16];
    float lmax = -3.4e38f;
    #pragma unroll
    for (int ch = 0; ch < 16; ++ch) {
        int j = ch * 256 + t;
        float pos_j = (float)token_positions[j];
        float dist = fabsf(pos_i - pos_j);
        int ph_j = photon[j];
        float rs = ph_j ? 1.0f : expf(-dist / c);
        float redshift = (i == j) ? 0.0f : rs;
        bool past = pos_j < pos_i;
        float vv = v_arr[j];
        bool causal = past && (vv >= 1e-6f) && (dist / fmaxf(vv, 1e-6f) <= c);
        float b = base[(size_t)i * LN + j];
        float br = b * redshift;
        float s = (ph_i && past) ? br : (causal ? br : 0.0f);
        vals[ch] = s;
        scores[(size_t)i * LN + j] = s;
        float ss = (s == 0.0f) ? NEG_BIG : s;
        lmax = fmaxf(lmax, ss);
    }
    // wave32 max-reduce, then cross-wave via LDS
    #pragma unroll
    for (int off = 16; off > 0; off >>= 1)
        lmax = fmaxf(lmax, __shfl_xor(lmax, off, 32));
    if ((t & 31) == 0) red[t >> 5] = lmax;
    __syncthreads();
    float M = red[0];
    #pragma unroll
    for (int w = 1; w < 8; ++w) M = fmaxf(M, red[w]);
    __syncthreads();

    float lsum = 0.0f;
    #pragma unroll
    for (int ch = 0; ch < 16; ++ch) {
        float s = vals[ch];
        float ss = (s == 0.0f) ? NEG_BIG : s;
        lsum += __expf(ss - M);
    }
    #pragma unroll
    for (int off = 16; off > 0; off >>= 1)
        lsum += __shfl_xor(lsum, off, 32);
    if ((t & 31) == 0) red[t >> 5] = lsum;
    __syncthreads();
    if (t == 0) {
        float S = 0.0f;
        #pragma unroll
        for (int w = 0; w < 8; ++w) S += red[w];
        rowmax[i] = M;
        rowsum[i] = S;
    }
}

// ---------------------------------------------------------------------------
// Kernel 3: V_red = token_vecs @ W_r^T + b_r  (M=4096, N=256, K=256), fp32 WMMA.
// One wave per 16x16 output tile. A 16x4 layout: lane m holds row m, lane half
// selects K pair {k,k+1} vs {k+2,k+3}. B[k][n] = W_r[n][k] -> contiguous loads.
// ---------------------------------------------------------------------------
__global__ void vred_kernel(const float* __restrict__ tv,
                            const float* __restrict__ Wr,
                            const float* __restrict__ br,
                            float* __restrict__ vred) {
    const int tile = blockIdx.x;               // (LN/16)*(DN/16)
    const int ti = tile / (DN / 16);
    const int tn = tile % (DN / 16);
    const int lane = threadIdx.x;
    const int m = lane & 15;
    const int hi = lane >> 4;
    const int arow = ti * 16 + m;              // A row for this lane
    const int bcol = tn * 16 + m;              // B column (n) for this lane

    v8f acc = {};
    #pragma unroll 4
    for (int k = 0; k < DN; k += 4) {
        int kk = k + 2 * hi;
        float2 av = *(const float2*)(tv + (size_t)arow * DN + kk);
        float2 bv = *(const float2*)(Wr + (size_t)bcol * DN + kk);
        v2f a; a.x = av.x; a.y = av.y;
        v2f b; b.x = bv.x; b.y = bv.y;
        acc = __builtin_amdgcn_wmma_f32_16x16x4_f32(false, a, false, b,
                                                    (short)0, acc, false, false);
    }
    float bias = br[tn * 16 + m];
    #pragma unroll
    for (int r = 0; r < 8; ++r) {
        int row = ti * 16 + r + 8 * hi;        // C/D layout: reg r, lane half
        vred[(size_t)row * DN + tn * 16 + m] = acc[r] + bias;
    }
}

// ---------------------------------------------------------------------------
// Kernel 4: attended = A @ V_red  (M=4096, N=256, K=4096), fp32 WMMA.
// One wave computes a 16x(16*NT) strip: A-fragment (and its 2 exps) built once
// per K-step and reused for NT WMMAs. A = (s==0)?0:__expf(s-max)/sum, scores
// stay L2-resident (64 MB < 192 MB L2).
// ---------------------------------------------------------------------------
__global__ void attend_kernel(const float* __restrict__ scores,
                              const float* __restrict__ vred,
                              const float* __restrict__ rowmax,
                              const float* __restrict__ rowsum,
                              float* __restrict__ out) {
    const int tile = blockIdx.x;               // (LN/16)*(DN/(16*NT))
    const int ti = tile / (DN / (16 * NT));
    const int tg = tile % (DN / (16 * NT));    // N group of NT tiles
    const int lane = threadIdx.x;
    const int m = lane & 15;
    const int hi = lane >> 4;
    const int arow = ti * 16 + m;
    const int ncol = tg * (16 * NT) + m;       // first N column for this lane

    const float mx = rowmax[arow];
    const float inv = 1.0f / rowsum[arow];
    const float* __restrict__ srow = scores + (size_t)arow * LN;

    v8f acc[NT];
    #pragma unroll
    for (int t = 0; t < NT; ++t) acc[t] = (v8f){};

    #pragma unroll 2
    for (int k = 0; k < LN; k += 4) {
        int kk = k + 2 * hi;
        float2 sc = *(const float2*)(srow + kk);
        v2f a;
        a.x = (sc.x == 0.0f) ? 0.0f : __expf(sc.x - mx) * inv;
        a.y = (sc.y == 0.0f) ? 0.0f : __expf(sc.y - mx) * inv;
        const float* b0 = vred + (size_t)kk * DN + ncol;
        const float* b1 = b0 + DN;
        #pragma unroll
        for (int t = 0; t < NT; ++t) {
            v2f b; b.x = b0[t * 16]; b.y = b1[t * 16];
            acc[t] = __builtin_amdgcn_wmma_f32_16x16x4_f32(false, a, false, b,
                                                           (short)0, acc[t],
                                                           false, false);
        }
    }
    #pragma unroll
    for (int t = 0; t < NT; ++t) {
        #pragma unroll
        for (int r = 0; r < 8; ++r) {
            int row = ti * 16 + r + 8 * hi;
            out[(size_t)row * DN + tg * (16 * NT) + t * 16 + m] = acc[t][r];
        }
    }
}

// ---------------------------------------------------------------------------
extern "C" void kernel_launch(void* const* d_in, const int* in_sizes, int n_in,
                              void* d_out, int out_size, void* d_ws, size_t ws_size,
                              hipStream_t stream) {
    const float* token_vecs      = (const float*)d_in[0];
    const int*   token_positions = (const int*)  d_in[1];
    const int*   pos_indices     = (const int*)  d_in[2];
    const float* base_scores     = (const float*)d_in[3];
    const float* log_c           = (const float*)d_in[4];
    const float* particle_mass   = (const float*)d_in[5];
    const float* W_r             = (const float*)d_in[6];
    const float* b_r             = (const float*)d_in[7];

    float* attended    = (float*)d_out;                         // L*D
    float* scores      = attended + (size_t)LN * DN;            // L*L
    float* photon_mask = scores + (size_t)LN * LN;              // L

    float* w      = (float*)d_ws;
    float* v_arr  = w;                       // L floats
    int*   photon = (int*)(w + LN);          // L ints
    float* rowmax = w + 2 * LN;              // L floats
    float* rowsum = w + 3 * LN;              // L floats
    float* vred   = w + 4 * LN;              // L*D floats (~4 MB)

    prep_kernel<<<LN / 256, 256, 0, stream>>>(pos_indices, particle_mass, log_c,
                                              photon, v_arr, photon_mask);
    vred_kernel<<<(LN / 16) * (DN / 16), 32, 0, stream>>>(token_vecs, W_r, b_r, vred);
    scores_kernel<<<LN, 256, 0, stream>>>(base_scores, token_positions, photon,
                                          v_arr, log_c, scores, rowmax, rowsum);
    attend_kernel<<<(LN / 16) * (DN / (16 * NT)), 32, 0, stream>>>(scores, vred,
                                                                   rowmax, rowsum,
                                                                   attended);
}